// DEMA_63264868270150
// MI455X (gfx1250) — compile-verified
//
#include <hip/hip_runtime.h>

// DEMA over time axis: x[B=32, L=4096, C=512] fp32 -> out same shape.
// Memory-bound (~512MB min traffic -> ~22us @ 23.3TB/s). Strategy:
//  - 16-timestep blocks of the EMA recurrence expressed as 16x16 lower-
//    triangular matmul, executed with V_WMMA_F32_16X16X4_F32 (K-accum x4 +
//    one rank-1 carry WMMA). Two chained EMAs per tile (10 WMMAs/tile).
//  - L split into 4 chunks of 1024 with 256-step zero-carry warmup
//    (0.9^256 ~ 2e-12 << fp32 eps) -> 4096 independent waves.
//  - y_0 = x_0 boundary handled exactly by seeding both carries with x_0.

typedef __attribute__((ext_vector_type(2))) float v2f;
typedef __attribute__((ext_vector_type(8))) float v8f;

#define EMA_A 0.1f   // alpha
#define EMA_B 0.9f   // 1 - alpha

static __device__ __forceinline__ v8f wmma4(v2f a, v2f b, v8f c) {
    // 8 args: (neg_a, A, neg_b, B, c_mod, C, reuse_a, reuse_b)
    return __builtin_amdgcn_wmma_f32_16x16x4_f32(false, a, false, b,
                                                 (short)0, c, false, false);
}

static __device__ __forceinline__ float p9(int e) {
    float r = 1.f;
    for (int i = 0; i < e; ++i) r *= EMA_B;
    return r;
}

__global__ __launch_bounds__(128) void dema_wmma_kernel(
    const float* __restrict__ x, float* __restrict__ out) {
    constexpr int L = 4096, C = 512;
    constexpr int CHUNK = 1024, WARM = 256;

    const int lane = threadIdx.x & 31;
    const int wave = threadIdx.x >> 5;
    const int m    = lane & 15;   // row (M) / column-in-group (N)
    const int half = lane >> 4;   // 0: lanes 0-15, 1: lanes 16-31

    const int bid   = blockIdx.x;
    const int cg    = bid & 7;          // 8 channel groups of 64
    const int chunk = (bid >> 3) & 3;   // 4 chunks of 1024 timesteps
    const int b     = bid >> 5;         // 32 batches

    const int col = cg * 64 + wave * 16 + m;

    // A-matrix K-slices: A[i][j] = alpha * 0.9^(i-j), lower triangular.
    // 32-bit A 16x4 layout: VGPR0 = {K0 lo-lanes, K2 hi-lanes},
    //                       VGPR1 = {K1 lo-lanes, K3 hi-lanes}.
    v2f A[4], Ac;
#pragma unroll
    for (int s = 0; s < 4; ++s) {
        const int jx = 4 * s + 2 * half;
        const int jy = jx + 1;
        A[s].x = (jx <= m) ? EMA_A * p9(m - jx) : 0.f;
        A[s].y = (jy <= m) ? EMA_A * p9(m - jy) : 0.f;
    }
    // Carry rank-1 term: d[i] = 0.9^(i+1) in K-row 0 only.
    Ac.x = (half == 0) ? p9(m + 1) : 0.f;
    Ac.y = 0.f;

    const int ts      = chunk * CHUNK;
    const int t0start = (chunk == 0) ? 0 : (ts - WARM);

    float y1p, y2p;  // carries for EMA1 / EMA2
    if (chunk == 0) {
        const float x0 = x[(size_t)b * L * C + col];
        y1p = x0;  // makes y_0 = 0.1*x_0 + 0.9*x_0 = x_0 exactly
        y2p = x0;
    } else {
        y1p = 0.f;
        y2p = 0.f;
    }

    const float* xb = x   + (size_t)b * L * C + col;
    float*       ob = out + (size_t)b * L * C + col;

    for (int t0 = t0start; t0 < ts + CHUNK; t0 += 16) {
        // Load X tile in B layout (4x16 slices; K = 4s + 2*half (+1)).
        const float* p = xb + (size_t)t0 * C;
        v2f bx[4];
#pragma unroll
        for (int s = 0; s < 4; ++s) {
            const int r = 4 * s + 2 * half;
            bx[s].x = p[(size_t)r * C];
            bx[s].y = p[(size_t)(r + 1) * C];
        }

        // ---- EMA1: Y = A*X + d (x) y1p ----
        v8f cy = {};
        v2f bc;
        bc.x = half ? 0.f : y1p;
        bc.y = 0.f;
        cy = wmma4(Ac, bc, cy);
#pragma unroll
        for (int s = 0; s < 4; ++s) cy = wmma4(A[s], bx[s], cy);

        // Cross-half copies for C->B relayout + carry extraction.
        float sx[8];
#pragma unroll
        for (int r = 0; r < 8; ++r) sx[r] = __shfl_xor(cy[r], 16, 32);
        y1p = half ? cy[7] : sx[7];  // Y[15][n]

        // Y (C layout: VGPRr = rows r / r+8) -> B layout slices.
        v2f yb[4];
        yb[0].x = half ? sx[2] : cy[0];  yb[0].y = half ? sx[3] : cy[1];
        yb[1].x = half ? sx[6] : cy[4];  yb[1].y = half ? sx[7] : cy[5];
        yb[2].x = half ? cy[2] : sx[0];  yb[2].y = half ? cy[3] : sx[1];
        yb[3].x = half ? cy[6] : sx[4];  yb[3].y = half ? cy[7] : sx[5];

        // ---- EMA2: Z = A*Y + d (x) y2p ----
        v8f cz = {};
        v2f bc2;
        bc2.x = half ? 0.f : y2p;
        bc2.y = 0.f;
        cz = wmma4(Ac, bc2, cz);
#pragma unroll
        for (int s = 0; s < 4; ++s) cz = wmma4(A[s], yb[s], cz);

        const float sz7 = __shfl_xor(cz[7], 16, 32);
        y2p = half ? cz[7] : sz7;  // Z[15][n]

        // ---- DEMA = 2Y - Z; store (skip warmup tiles) ----
        if (t0 >= ts) {
            float* q = ob + (size_t)(t0 + 8 * half) * C;
#pragma unroll
            for (int r = 0; r < 8; ++r)
                __builtin_nontemporal_store(2.f * cy[r] - cz[r],
                                            &q[(size_t)r * C]);
        }
    }
}

extern "C" void kernel_launch(void* const* d_in, const int* in_sizes, int n_in,
                              void* d_out, int out_size, void* d_ws,
                              size_t ws_size, hipStream_t stream) {
    const float* x = (const float*)d_in[0];
    float* out = (float*)d_out;
    // 32 batches x 4 L-chunks x 8 channel-groups = 1024 blocks, 128 thr (4 waves)
    dema_wmma_kernel<<<dim3(1024), dim3(128), 0, stream>>>(x, out);
}